// MyIN_60181081752397
// MI455X (gfx1250) — compile-verified
//
#include <hip/hip_runtime.h>
#include <hip/hip_bf16.h>

// ---------------------------------------------------------------------------
// SPADE-like fused affine:
//   out[b,c,h,w] = x[b,c,h,w] * scale[b,h,w] + shift[b,h,w]
//   scale = sum(wr) + sum_k (gamma[b,k]*wr[k]) * sem[b,k,h,w]
//   shift = br     + sum_k (beta [b,k]*wr[k]) * sem[b,k,h,w]
// Memory-bound: ~69 MB HBM traffic -> ~3 us at 23.3 TB/s. Kernel A (tiny,
// WMMA f32 16x16x4 for the 4x256x128 GEMM) precomputes the 4x12 coefficient
// table; kernel B is a pure b128 streaming pass with NT hints.
// ---------------------------------------------------------------------------

#define BATCH    4
#define CCH      128
#define HW       16384        // 128*128
#define LANG_DIM 256
#define NHIDDEN  128
#define NUM_OP   6
#define SEM_CH   8            // input_semantics channels; we use 2..7

typedef float v2f __attribute__((ext_vector_type(2)));
typedef float v8f __attribute__((ext_vector_type(8)));
typedef float f4  __attribute__((ext_vector_type(4)));

// ------------------------- Kernel A: coefficients ---------------------------
// One 256-thread block = 8 waves. Wave w computes actv[:, 16w : 16w+16] via
// V_WMMA_F32_16X16X4_F32 (M=16 padded from 4, K=256 in 64 steps of 4).
// A-row padding is done branch-free: clamp row index (always-valid address,
// straight b64 loads) and multiply by a 0/1 lane mask — no EXEC juggling,
// which WMMA requires to be all-1s anyway.
__global__ __launch_bounds__(256) void spade_coef_kernel(
    const float* __restrict__ lang, const float* __restrict__ Ws,
    const float* __restrict__ bs,   const float* __restrict__ Wg,
    const float* __restrict__ bg,   const float* __restrict__ Wb,
    const float* __restrict__ bb,   const float* __restrict__ Wr,
    const float* __restrict__ br,   float* __restrict__ coef)
{
    __shared__ float actv[BATCH * NHIDDEN];   // [b][n]

    const int tid  = threadIdx.x;
    const int lane = tid & 31;
    const int wave = tid >> 5;          // 0..7 -> N tile
    const int n0   = wave * 16;
    const int m    = lane & 15;         // A row / B column within tile
    const int hi   = lane >> 4;         // K half selector (lanes 16-31: K+2)

    // A: lang_embed (4x256), rows 4..15 zero-padded via mask.
    // B: Ws.T -> B[k][n] = Ws[n*256 + k].
    const float amask = (m < 4) ? 1.0f : 0.0f;
    const v2f* arow = (const v2f*)(lang + (size_t)(m & 3) * LANG_DIM);
    const v2f* brow = (const v2f*)(Ws   + (size_t)(n0 + m) * LANG_DIM);

    v8f c = {};
    #pragma unroll 4
    for (int k0 = 0; k0 < LANG_DIM; k0 += 4) {
        const int kp = (k0 >> 1) + hi;  // v2f index of {K, K+1} pair
        v2f a  = arow[kp];              // global_load_b64
        v2f bv = brow[kp];              // global_load_b64
        a.x *= amask;
        a.y *= amask;
        // (neg_a, A, neg_b, B, c_mod, C, reuse_a, reuse_b)
        c = __builtin_amdgcn_wmma_f32_16x16x4_f32(
                false, a, false, bv, (short)0, c, false, false);
    }

    // D layout: VGPR j, lanes 0-15 -> M=j. Valid batch rows are M=0..3.
    if (lane < 16) {
        const float bias = bs[n0 + lane];
        #pragma unroll
        for (int j = 0; j < 4; ++j)
            actv[j * NHIDDEN + n0 + lane] = c[j] + bias;
    }
    __syncthreads();

    // gamma/beta heads (4 batches x 6 ops) + wr fold.
    if (tid < BATCH * NUM_OP) {
        const int b = tid / NUM_OP, k = tid % NUM_OP;
        const float* a  = actv + b * NHIDDEN;
        const float* wg = Wg + k * NHIDDEN;
        const float* wb = Wb + k * NHIDDEN;
        float g = bg[k], be = bb[k];
        #pragma unroll 8
        for (int h = 0; h < NHIDDEN; ++h) {
            const float av = a[h];
            g  = fmaf(av, wg[h], g);
            be = fmaf(av, wb[h], be);
        }
        const float w = Wr[k];
        coef[b * 12 + k]          = g  * w;   // gw
        coef[b * 12 + NUM_OP + k] = be * w;   // bw
    } else if (tid == BATCH * NUM_OP) {
        float s = 0.0f;
        #pragma unroll
        for (int k = 0; k < NUM_OP; ++k) s += Wr[k];
        coef[48] = s;        // sum(wr)
        coef[49] = br[0];
    }
}

// ------------------------- Kernel B: streaming pass -------------------------
// grid = (HW/4/256, BATCH*4). blockIdx.y encodes (batch, 32-channel chunk) so
// coefficient loads are wave-uniform (scalar). Each thread: 4 pixels, 6 sem
// f4 loads -> scale/shift, then 32 channel iterations of b128 load/FMA/store
// with non-temporal hints (single-use streaming data).
__global__ __launch_bounds__(256) void spade_apply_kernel(
    const float* __restrict__ x, const float* __restrict__ sem_in,
    const float* __restrict__ coef, float* __restrict__ out)
{
    const int p4 = blockIdx.x * 256 + threadIdx.x;   // float4 spatial index
    const int b  = blockIdx.y >> 2;
    const int cc = blockIdx.y & 3;                   // channel chunk (32 ch)

    const float* cb    = coef + b * 12;              // uniform -> s_load
    const float  wrsum = coef[48];
    const float  br0   = coef[49];

    // semantics channels 2..7
    const f4* sem = (const f4*)(sem_in + (size_t)(b * SEM_CH + 2) * HW) + p4;
    f4 scale = {wrsum, wrsum, wrsum, wrsum};
    f4 shift = {br0,   br0,   br0,   br0};
    #pragma unroll
    for (int k = 0; k < NUM_OP; ++k) {
        const f4 s = sem[k * (HW / 4)];              // L2-resident, reused
        scale += cb[k]          * s;
        shift += cb[NUM_OP + k] * s;
    }

    const size_t base = ((size_t)b * CCH + (size_t)cc * 32) * HW;
    const f4* xp = (const f4*)(x   + base) + p4;
    f4*       op = (f4*)      (out + base) + p4;
    #pragma unroll 4
    for (int c = 0; c < 32; ++c) {
        f4 v = __builtin_nontemporal_load(xp + (size_t)c * (HW / 4));
        v = v * scale + shift;
        __builtin_nontemporal_store(v, op + (size_t)c * (HW / 4));
    }
}

// ---------------------------------------------------------------------------
extern "C" void kernel_launch(void* const* d_in, const int* in_sizes, int n_in,
                              void* d_out, int out_size, void* d_ws, size_t ws_size,
                              hipStream_t stream) {
    const float* x    = (const float*)d_in[0];
    const float* lang = (const float*)d_in[1];
    const float* sem  = (const float*)d_in[2];
    const float* Ws   = (const float*)d_in[3];
    const float* bs   = (const float*)d_in[4];
    const float* Wg   = (const float*)d_in[5];
    const float* bg   = (const float*)d_in[6];
    const float* Wb   = (const float*)d_in[7];
    const float* bb   = (const float*)d_in[8];
    const float* Wr   = (const float*)d_in[9];
    const float* br   = (const float*)d_in[10];
    float* out  = (float*)d_out;
    float* coef = (float*)d_ws;   // 50 floats

    spade_coef_kernel<<<1, 256, 0, stream>>>(lang, Ws, bs, Wg, bg, Wb, bb,
                                             Wr, br, coef);

    dim3 grid(HW / 4 / 256, BATCH * 4);   // (16, 16)
    spade_apply_kernel<<<grid, 256, 0, stream>>>(x, sem, coef, out);
}